// Ridge_55216099558209
// MI455X (gfx1250) — compile-verified
//
#include <hip/hip_runtime.h>

// Blocked recursive-least-squares (Woodbury) formulation of prefix ridge
// regression. One workgroup (1 wave, 32 threads) per batch element.
// Matmuls routed through V_WMMA_F32_16X16X4_F32 (CDNA5 fp32 tensor path).
// Chunk staging via the Tensor Data Mover (tensor_load_to_lds + TENSORcnt).

typedef __attribute__((ext_vector_type(2))) float v2f;
typedef __attribute__((ext_vector_type(8))) float v8f;
typedef __attribute__((ext_vector_type(4))) unsigned int v4u;
typedef __attribute__((ext_vector_type(8))) int v8i_t;
typedef __attribute__((ext_vector_type(4))) int v4i_t;

#define DDIM 64
#define NSTEP 128
#define CK 16
#define NCHUNK (NSTEP / CK)
#define USE_TDM 1

__device__ __forceinline__ v8f wmma_f32(v2f a, v2f b, v8f c) {
    // 8 args: (neg_a, A, neg_b, B, c_mod, C, reuse_a, reuse_b)
    // f32 WMMA: A/B negation must be 0 per ISA (NEG = CNeg only).
    return __builtin_amdgcn_wmma_f32_16x16x4_f32(false, a, false, b,
                                                 (short)0, c, false, false);
}

__global__ __launch_bounds__(32) void rls_wmma_kernel(
    const float* __restrict__ data,     // [B, N, D]
    const float* __restrict__ targets,  // [B, N]
    float* __restrict__ out) {          // [B, N]
    __shared__ __align__(16) float sP[DDIM * DDIM];  // (lam*I + G)^-1, row-major
    __shared__ __align__(16) float sX[CK * DDIM];    // current chunk rows
    __shared__ __align__(16) float sU[CK * DDIM];    // U = X * P
    __shared__ __align__(16) float sT[CK * DDIM];    // T = (-Btilde) * U
    __shared__ float sS[CK * CK];                    // S = X P X^T
    __shared__ float sB[CK * CK];                    // Btilde accumulator
    __shared__ float sw[DDIM];                       // w = P * H
    __shared__ float sa[CK];
    __shared__ float sg[CK];
    __shared__ float sp0[CK];
    __shared__ float sy[CK];
    __shared__ float sscal[2];

    const int b   = blockIdx.x;
    const int tid = threadIdx.x;
    const int lm  = tid & 15;        // M/N index inside a 16-wide tile
    const int hi  = tid >> 4;        // half-wave select
    const int hk  = hi * 2;          // K-pair base for A/B operands

    // P0 = (1/lam) I with lam = 1 ; w0 = 0
    for (int i = tid; i < DDIM * DDIM; i += 32)
        sP[i] = ((i / DDIM) == (i % DDIM)) ? 1.0f : 0.0f;
    for (int i = tid; i < DDIM; i += 32) sw[i] = 0.0f;

    const float* Xg   = data + (size_t)b * NSTEP * DDIM;
    const float* yg   = targets + (size_t)b * NSTEP;
    float*       outg = out + (size_t)b * NSTEP;

#if USE_TDM
    // Invariant parts of the Tensor DMA descriptor (D#), ISA 8.3/8.4:
    // 2D tensor, data_size=4B, tensor_dim0=64, tensor_dim1=16 (one chunk),
    // tile_dim0=64, tile_dim1=16, dim0 stride=64 elements. Groups 2/3 = 0
    // (<=2D tensor). LDS destination = sX (LDS offset = low 32 addr bits).
    const unsigned int lds_off = (unsigned int)(unsigned long long)(uintptr_t)sX;
    const v8i_t g1 = { (int)0x00020000u,        // workgroup_mask=0, data_size=2 (4B)
                       (int)(64u << 16),        // tensor_dim0 = 64
                       (int)(16u << 16),        // tensor_dim1 = 16
                       (int)(64u << 16),        // tile_dim0 = 64
                       (int)16u,                // tile_dim1 = 16, tile_dim2 = 0
                       (int)64u,                // tensor_dim0_stride = 64
                       0, 0 };                  // dim1 stride unused (2D tile)
    const v4i_t gz4 = { 0, 0, 0, 0 };
    const v8i_t gz8 = { 0, 0, 0, 0, 0, 0, 0, 0 };
#endif
    __syncthreads();

    for (int c = 0; c < NCHUNK; ++c) {
        // ---- stage chunk X (16x64) and y (16) into LDS ----
#if USE_TDM
        {
            unsigned long long ga =
                (unsigned long long)(uintptr_t)(Xg + c * CK * DDIM);
            v4u g0 = { 1u,                                   // count=1, no gather
                       lds_off,                              // lds_addr
                       (unsigned int)ga,                     // global_addr[31:0]
                       (unsigned int)((ga >> 32) & 0x1FFFFFFull)
                           | 0x80000000u };                  // addr[56:32], type=2
            // 6-arg form (clang-23 / therock-10.0 headers):
            // (group0, group1, group2, group3, group_pad, cpol)
            __builtin_amdgcn_tensor_load_to_lds(g0, g1, gz4, gz4, gz8, 0);
        }
        if (tid < CK) sy[tid] = yg[c * CK + tid];
        if (c + 1 < NCHUNK)  // warm L2 for next chunk (global_prefetch_b8)
            __builtin_prefetch(Xg + (c + 1) * CK * DDIM + tid * 32, 0, 1);
        __builtin_amdgcn_s_wait_tensorcnt(0);
#else
        {
            const float4* src = (const float4*)(Xg + c * CK * DDIM);
            float4*       dst = (float4*)sX;
            for (int i = tid; i < CK * DDIM / 4; i += 32) dst[i] = src[i];
            if (tid < CK) sy[tid] = yg[c * CK + tid];
            if (c + 1 < NCHUNK)
                __builtin_prefetch(Xg + (c + 1) * CK * DDIM + tid * 32, 0, 1);
        }
#endif
        __syncthreads();

        // ---- U = X * P  (16x64): 4 output tiles, K = 64 in steps of 4 ----
        for (int t = 0; t < 4; ++t) {
            v8f acc = {};
            for (int kk = 0; kk < 16; ++kk) {
                v2f a, bm;
                a.x  = sX[lm * DDIM + 4 * kk + hk];
                a.y  = sX[lm * DDIM + 4 * kk + hk + 1];
                bm.x = sP[(4 * kk + hk) * DDIM + t * 16 + lm];
                bm.y = sP[(4 * kk + hk + 1) * DDIM + t * 16 + lm];
                acc  = wmma_f32(a, bm, acc);
            }
            for (int v = 0; v < 8; ++v)
                sU[(v + 8 * hi) * DDIM + t * 16 + lm] = acc[v];
        }
        __syncthreads();

        // ---- S = U * X^T (16x16) ----
        {
            v8f acc = {};
            for (int kk = 0; kk < 16; ++kk) {
                v2f a, bm;
                a.x  = sU[lm * DDIM + 4 * kk + hk];
                a.y  = sU[lm * DDIM + 4 * kk + hk + 1];
                bm.x = sX[lm * DDIM + 4 * kk + hk];   // (X^T)[k][n] = X[n][k]
                bm.y = sX[lm * DDIM + 4 * kk + hk + 1];
                acc  = wmma_f32(a, bm, acc);
            }
            for (int v = 0; v < 8; ++v)
                sS[(v + 8 * hi) * CK + lm] = acc[v];
        }
        // p0 = X * w ; zero a, Btilde
        if (tid < CK) {
            float s = 0.0f;
            for (int d = 0; d < DDIM; ++d) s += sX[tid * DDIM + d] * sw[d];
            sp0[tid] = s;
            sa[tid]  = 0.0f;
        }
        for (int i = tid; i < CK * CK; i += 32) sB[i] = 0.0f;
        __syncthreads();

        // ---- 16 sequential rank-1 steps, all in the 16-dim chunk subspace ----
        for (int j = 0; j < CK; ++j) {
            if (tid < CK) {  // g = e_j - Btilde * S[:,j]
                float g = (tid == j) ? 1.0f : 0.0f;
                for (int u = 0; u < CK; ++u)
                    g -= sB[tid * CK + u] * sS[u * CK + j];
                sg[tid] = g;
            }
            __syncthreads();
            if (tid == 0) {
                float dsum = 0.0f, psum = sp0[j];
                for (int u = 0; u < CK; ++u) {
                    float su = sS[u * CK + j];
                    dsum += su * sg[u];
                    psum += su * sa[u];
                }
                float alpha = 1.0f / (1.0f + dsum);
                float err   = sy[j] - psum;
                outg[c * CK + j] = psum;  // pred BEFORE absorbing sample j
                sscal[0] = alpha * err;
                sscal[1] = alpha;
            }
            __syncthreads();
            if (tid < CK) {  // a += alpha*err*g ; Btilde += alpha * g g^T
                float ae = sscal[0], al = sscal[1];
                float gt = sg[tid];
                sa[tid] += ae * gt;
                for (int u = 0; u < CK; ++u)
                    sB[tid * CK + u] += al * gt * sg[u];
            }
            __syncthreads();
        }

        // ---- fold chunk back into full state ----
        // w += U^T a
        for (int d = tid; d < DDIM; d += 32) {
            float s = sw[d];
            for (int j2 = 0; j2 < CK; ++j2) s += sU[j2 * DDIM + d] * sa[j2];
            sw[d] = s;
        }
        // negate Btilde so the P update can accumulate in WMMA's C operand
        for (int i = tid; i < CK * CK; i += 32) sB[i] = -sB[i];
        __syncthreads();

        // T = (-Btilde) * U   (16x64)
        for (int t = 0; t < 4; ++t) {
            v8f acc = {};
            for (int kk = 0; kk < 4; ++kk) {
                v2f a, bm;
                a.x  = sB[lm * CK + 4 * kk + hk];
                a.y  = sB[lm * CK + 4 * kk + hk + 1];
                bm.x = sU[(4 * kk + hk) * DDIM + t * 16 + lm];
                bm.y = sU[(4 * kk + hk + 1) * DDIM + t * 16 + lm];
                acc  = wmma_f32(a, bm, acc);
            }
            for (int v = 0; v < 8; ++v)
                sT[(v + 8 * hi) * DDIM + t * 16 + lm] = acc[v];
        }
        __syncthreads();

        // P += U^T * T  (== P - U^T Btilde U), C preloaded from P tile
        for (int r = 0; r < 4; ++r) {
            for (int t = 0; t < 4; ++t) {
                v8f acc;
                for (int v = 0; v < 8; ++v)
                    acc[v] = sP[(r * 16 + v + 8 * hi) * DDIM + t * 16 + lm];
                for (int kk = 0; kk < 4; ++kk) {
                    v2f a, bm;
                    a.x  = sU[(4 * kk + hk) * DDIM + r * 16 + lm];  // (U^T) tile
                    a.y  = sU[(4 * kk + hk + 1) * DDIM + r * 16 + lm];
                    bm.x = sT[(4 * kk + hk) * DDIM + t * 16 + lm];
                    bm.y = sT[(4 * kk + hk + 1) * DDIM + t * 16 + lm];
                    acc  = wmma_f32(a, bm, acc);
                }
                for (int v = 0; v < 8; ++v)
                    sP[(r * 16 + v + 8 * hi) * DDIM + t * 16 + lm] = acc[v];
            }
        }
        __syncthreads();
    }
}

extern "C" void kernel_launch(void* const* d_in, const int* in_sizes, int n_in,
                              void* d_out, int out_size, void* d_ws, size_t ws_size,
                              hipStream_t stream) {
    const float* data    = (const float*)d_in[0];
    const float* targets = (const float*)d_in[1];
    float*       out     = (float*)d_out;
    const int B = in_sizes[1] / NSTEP;  // targets is [B, N]
    hipLaunchKernelGGL(rls_wmma_kernel, dim3(B), dim3(32), 0, stream,
                       data, targets, out);
}